// GraphConvolutionalEncoder_14035953123780
// MI455X (gfx1250) — compile-verified
//
#include <hip/hip_runtime.h>

// GCN 2-layer forward for MI455X (gfx1250, wave32).
// Memory-bound: ~1.5 GB of gather/scatter traffic (~65us @ 23.3 TB/s) dominates;
// GEMMs (~2.4 GFLOP total) use fp32 WMMA (V_WMMA_F32_16X16X4_F32) for exact
// fp32 math on the matrix pipe. One wave computes a full 16xN row-block so each
// A fragment is loaded once and reused across N/16 accumulators; all W/X loads
// use compile-time immediate offsets (no 64-bit address math in the hot loop).
// Layer-1 ReLU is fused into layer-2's GEMM A-fragment loads.

#define N_NODES 50000   // divisible by 16 (3125 row-blocks)
#define N_EDGES 800000
#define DIM_IN  128
#define DIM_HID 128
#define DIM_OUT 64

typedef __attribute__((ext_vector_type(2))) float v2f;
typedef __attribute__((ext_vector_type(8))) float v8f;

// ---------------- degree / dinv ----------------

__global__ void init_deg_kernel(float* __restrict__ deg, int n) {
    int i = blockIdx.x * blockDim.x + threadIdx.x;
    if (i < n) deg[i] = 1.0f;   // self-loop
}

__global__ void count_deg_kernel(const long long* __restrict__ dst,
                                 float* __restrict__ deg, int e) {
    int i = blockIdx.x * blockDim.x + threadIdx.x;
    if (i < e) atomicAdd(&deg[(int)dst[i]], 1.0f);
}

__global__ void rsqrt_kernel(float* __restrict__ deg, int n) {
    int i = blockIdx.x * blockDim.x + threadIdx.x;
    if (i < n) deg[i] = rsqrtf(deg[i]);
}

// ---------------- fp32 WMMA GEMM: support = act(X)(MxK) @ W(KxN) -----------
// One wave owns a 16-row block and all NT = N/16 column tiles:
//   - A fragment (16x4 f32, ISA 7.12.2 layout) loaded once per k-step,
//     reused by NT WMMAs into independent accumulators (no RAW chains).
//   - B/X addressed via a single base + compile-time immediate offsets.
// Epilogue also emits aggInit = support * dinv[m]^2 + bias[n] (self-loop+bias).

template <int K, int N, bool RELU_IN>
__global__ void gemm_wmma_f32_kernel(const float* __restrict__ X,
                                     const float* __restrict__ W,
                                     const float* __restrict__ bias,
                                     const float* __restrict__ dinv,
                                     float* __restrict__ support,
                                     float* __restrict__ aggInit,
                                     int M) {
    constexpr int NT = N / 16;           // 8 (layer 1) or 4 (layer 2)
    const int lane = threadIdx.x & 31;
    const int wave = threadIdx.x >> 5;
    const int wavesPerBlock = blockDim.x >> 5;
    const int rowBlk = blockIdx.x * wavesPerBlock + wave;
    if (rowBlk >= (M >> 4)) return;      // wave-uniform exit: EXEC stays all-1s

    const int m0  = rowBlk << 4;
    const int hi  = lane >> 4;           // 0: lanes 0-15, 1: lanes 16-31
    const int l15 = lane & 15;

    v8f acc[NT] = {};

    // All hot-loop offsets below are compile-time constants from these bases.
    const float* arow  = X + (m0 + l15) * K + 2 * hi;   // A: row m0+l15, K-pair
    const float* bbase = W + 2 * hi * N + l15;          // B: rows {2hi, 2hi+1}

#pragma unroll
    for (int k0 = 0; k0 < K; k0 += 4) {
        // A fragment: 2 contiguous K values for this half-wave (8B-aligned)
        v2f a = *(const v2f*)(arow + k0);
        if (RELU_IN) {                    // fused activation of previous layer
            a.x = fmaxf(a.x, 0.0f);
            a.y = fmaxf(a.y, 0.0f);
        }
#pragma unroll
        for (int t = 0; t < NT; ++t) {
            v2f b;
            b.x = bbase[k0 * N + t * 16];        // row k0+2hi   (imm offset)
            b.y = bbase[k0 * N + N + t * 16];    // row k0+2hi+1 (imm offset)
            // D = A*B + C  (8 args: neg_a, A, neg_b, B, c_mod, C, reuse_a/b)
            acc[t] = __builtin_amdgcn_wmma_f32_16x16x4_f32(
                false, a, false, b, (short)0, acc[t], false, false);
        }
    }

    // C/D layout: VGPR r, lanes 0-15 -> M=m0+r; lanes 16-31 -> M=m0+r+8
    float bn[NT];
#pragma unroll
    for (int t = 0; t < NT; ++t) bn[t] = bias[t * 16 + l15];

#pragma unroll
    for (int r = 0; r < 8; ++r) {
        const int   m   = m0 + r + 8 * hi;
        const float di  = dinv[m];
        const float di2 = di * di;
        float* srow = support + (size_t)m * N + l15;
        float* grow = aggInit + (size_t)m * N + l15;
#pragma unroll
        for (int t = 0; t < NT; ++t) {
            const float s = acc[t][r];
            srow[t * 16] = s;
            grow[t * 16] = s * di2 + bn[t];
        }
    }
}

// ---------------- edge scatter-add (bandwidth critical) ----------------
// One wave per edge; each lane owns VEC=D/32 contiguous features: one
// global_load_b128/b64 gather (hits L2: support fits in 192 MB) and VEC
// lane-contiguous global_atomic_add_f32 that coalesce into cacheline bursts.

template <int D>
__global__ void edge_scatter_kernel(const long long* __restrict__ src,
                                    const long long* __restrict__ dst,
                                    const float* __restrict__ dinv,
                                    const float* __restrict__ support,
                                    float* __restrict__ agg,
                                    int E) {
    constexpr int VEC = D / 32;          // 4 (D=128) or 2 (D=64)
    const int lane = threadIdx.x & 31;
    const int e = (int)((blockIdx.x * (unsigned)blockDim.x + threadIdx.x) >> 5);
    if (e >= E) return;                  // wave-uniform

    const int s = (int)src[e];
    const int d = (int)dst[e];
    const float coef = dinv[s] * dinv[d];

    const float* srow = support + (size_t)s * D + lane * VEC;
    float*       drow = agg     + (size_t)d * D + lane * VEC;

    float v[VEC];
#pragma unroll
    for (int j = 0; j < VEC; ++j) v[j] = srow[j];   // merges to b128/b64 load
#pragma unroll
    for (int j = 0; j < VEC; ++j) atomicAdd(&drow[j], v[j] * coef);
}

// ---------------- elementwise ReLU (final output only) ----------------

__global__ void relu_kernel(float* __restrict__ x, size_t n) {
    size_t i = (size_t)blockIdx.x * blockDim.x + threadIdx.x;
    if (i < n) x[i] = fmaxf(x[i], 0.0f);
}

// ---------------- launcher ----------------

extern "C" void kernel_launch(void* const* d_in, const int* in_sizes, int n_in,
                              void* d_out, int out_size, void* d_ws, size_t ws_size,
                              hipStream_t stream) {
    const float*     x      = (const float*)d_in[0];
    const long long* edges  = (const long long*)d_in[1];   // int64 [2, E]
    const float*     W1     = (const float*)d_in[2];
    const float*     b1     = (const float*)d_in[3];
    const float*     W2     = (const float*)d_in[4];
    const float*     b2     = (const float*)d_in[5];
    float*           out    = (float*)d_out;

    const long long* srcIdx = edges;              // edges[0]
    const long long* dstIdx = edges + N_EDGES;    // edges[1]

    // workspace layout (floats): dinv | support1 | agg1(pre-relu h1) | support2
    float* ws       = (float*)d_ws;
    float* dinv     = ws;
    float* support1 = ws + 50048;                                   // padded
    float* agg1     = support1 + (size_t)N_NODES * DIM_HID;
    float* support2 = agg1     + (size_t)N_NODES * DIM_HID;

    const int B = 256;
    const int wavesPerBlock = B / 32;
    const int rowBlks = N_NODES / 16;                                // 3125

    // degree -> dinv
    init_deg_kernel<<<(N_NODES + B - 1) / B, B, 0, stream>>>(dinv, N_NODES);
    count_deg_kernel<<<(N_EDGES + B - 1) / B, B, 0, stream>>>(dstIdx, dinv, N_EDGES);
    rsqrt_kernel<<<(N_NODES + B - 1) / B, B, 0, stream>>>(dinv, N_NODES);

    // ---- layer 1: support1 = x@W1 ; agg1 = support1*dinv^2 + b1 + scatter ----
    {
        gemm_wmma_f32_kernel<DIM_IN, DIM_HID, false>
            <<<(rowBlks + wavesPerBlock - 1) / wavesPerBlock, B, 0, stream>>>(
                x, W1, b1, dinv, support1, agg1, N_NODES);

        const long long thr = (long long)N_EDGES * 32;
        edge_scatter_kernel<DIM_HID><<<(unsigned)((thr + B - 1) / B), B, 0, stream>>>(
            srcIdx, dstIdx, dinv, support1, agg1, N_EDGES);
        // NOTE: no ReLU pass here -- fused into layer-2 GEMM A-loads.
    }

    // ---- layer 2: support2 = relu(agg1)@W2 ; out = ... ; final relu ----
    {
        gemm_wmma_f32_kernel<DIM_HID, DIM_OUT, true>
            <<<(rowBlks + wavesPerBlock - 1) / wavesPerBlock, B, 0, stream>>>(
                agg1, W2, b2, dinv, support2, out, N_NODES);

        const long long thr = (long long)N_EDGES * 32;
        edge_scatter_kernel<DIM_OUT><<<(unsigned)((thr + B - 1) / B), B, 0, stream>>>(
            srcIdx, dstIdx, dinv, support2, out, N_EDGES);

        const size_t n2 = (size_t)N_NODES * DIM_OUT;
        relu_kernel<<<(unsigned)((n2 + B - 1) / B), B, 0, stream>>>(out, n2);
    }
}